// LSTMGNN_27977416966548
// MI455X (gfx1250) — compile-verified
//
#include <hip/hip_runtime.h>
#include <math.h>

#define D 64
#define BB 64
#define LL 200

typedef __attribute__((ext_vector_type(2))) float v2f;
typedef __attribute__((ext_vector_type(8))) float v8f;

__device__ __forceinline__ float wave_reduce_sum(float v) {
#pragma unroll
  for (int m = 16; m >= 1; m >>= 1) v += __shfl_xor(v, m, 32);
  return v;
}
__device__ __forceinline__ float wave_reduce_max(float v) {
#pragma unroll
  for (int m = 16; m >= 1; m >>= 1) v = fmaxf(v, __shfl_xor(v, m, 32));
  return v;
}
__device__ __forceinline__ float fast_sigmoid(float x) {
  return __builtin_amdgcn_rcpf(1.0f + __expf(-x));  // v_rcp_f32
}

// e2 = emb * sigmoid(emb @ w0 + b0); acc2 = e2   (and same for w1 -> e3/acc3)
// One wave computes a 16-row strip via V_WMMA_F32_16X16X4_F32.
__global__ void gate_kernel(const float* __restrict__ emb,
                            const float* __restrict__ w0, const float* __restrict__ b0,
                            const float* __restrict__ w1, const float* __restrict__ b1,
                            float* __restrict__ e2, float* __restrict__ acc2,
                            float* __restrict__ e3, float* __restrict__ acc3,
                            int n) {
  const int lane = threadIdx.x & 31;
  const int wave = threadIdx.x >> 5;
  const int row0 = (blockIdx.x * 8 + wave) * 16;
  if (row0 >= n) return;  // wave-uniform: EXEC stays all-ones for WMMA
  const int lrow = lane & 15;
  const int lhi  = lane >> 4;  // 0 or 1

  int arow = row0 + lrow;
  if (arow >= n) arow = n - 1;  // clamp loads for partial tail strip

  // A panel: 16 rows x 64 cols held as 16 k-fragments of v2f.
  // A-frag layout (16x4 f32): lanes 0-15 hold K={0,1}, lanes 16-31 hold K={2,3}.
  v2f a[16];
#pragma unroll
  for (int kk = 0; kk < 16; ++kk) {
    const float* p = emb + (size_t)arow * D + 4 * kk + 2 * lhi;
    v2f t = {p[0], p[1]};
    a[kk] = t;
  }

  const float* Ws[2]   = {w0, w1};
  const float* Bias[2] = {b0, b1};
  float* Eo[2] = {e2, e3};
  float* Ao[2] = {acc2, acc3};

  for (int m = 0; m < 2; ++m) {
    const float* W = Ws[m];
    const float* bias = Bias[m];
    float* E  = Eo[m];
    float* AC = Ao[m];
#pragma unroll
    for (int nt = 0; nt < 4; ++nt) {
      const int col = nt * 16 + lrow;  // global output column for this lane
      v8f c = {};
#pragma unroll
      for (int kk = 0; kk < 16; ++kk) {
        const int k = 4 * kk + 2 * lhi;
        // B-frag layout (4x16 f32): lanes 0-15 hold K={0,1}, lanes 16-31 K={2,3}
        v2f bfrag = {W[k * D + col], W[(k + 1) * D + col]};
        c = __builtin_amdgcn_wmma_f32_16x16x4_f32(
            /*neg_a=*/false, a[kk], /*neg_b=*/false, bfrag,
            /*c_mod=*/(short)0, c, /*reuse_a=*/false, /*reuse_b=*/false);
      }
      const float bv = bias[col];
      // C/D layout: VGPR v, lanes 0-15 -> M=v; lanes 16-31 -> M=v+8; N=lane%16
#pragma unroll
      for (int v = 0; v < 8; ++v) {
        const int row = row0 + v + 8 * lhi;
        if (row < n) {
          const size_t off = (size_t)row * D + col;
          const float g = fast_sigmoid(c[v] + bv);
          const float val = emb[off] * g;
          E[off]  = val;
          AC[off] = val;
        }
      }
    }
  }
}

// y[idx0[e]] += val[e] * x[idx1[e]]
// One wave owns 32 edges: coalesced metadata (lane e -> edge base+e), then a
// broadcast loop; payload moved as float2 (global_load_b64) + 2 f32 atomics.
__global__ void spmm_kernel(const int* __restrict__ idx, const float* __restrict__ val,
                            const float* __restrict__ x, float* __restrict__ y,
                            int nnz) {
  const int lane = threadIdx.x & 31;
  const long long base =
      ((long long)blockIdx.x * 8 + (threadIdx.x >> 5)) * 32;
  if (base >= nnz) return;
  const long long e = base + lane;
  int dst = 0, src = 0;
  float v = 0.f;
  if (e < nnz) {
    dst = idx[e];
    src = idx[(size_t)nnz + e];
    v   = val[e];
  }
  const long long rem = (long long)nnz - base;
  const int cnt = rem < 32 ? (int)rem : 32;
  for (int i = 0; i < cnt; ++i) {
    const int s_   = __shfl(src, i, 32);
    const int d_   = __shfl(dst, i, 32);
    const float vv = __shfl(v, i, 32);
    const float2 xv = ((const float2*)(x + (size_t)s_ * D))[lane];
    float* yd = y + (size_t)d_ * D + 2 * lane;
    atomicAdd(yd,     vv * xv.x);
    atomicAdd(yd + 1, vv * xv.y);
  }
}

// acc += tmp / max(||tmp||, 1e-12);  ebuf = tmp   (one wave per row, float2/lane)
__global__ void norm_acc_kernel(const float* __restrict__ tmp, float* __restrict__ acc,
                                float* __restrict__ ebuf, int n) {
  const int lane = threadIdx.x & 31;
  const int row = blockIdx.x * 8 + (threadIdx.x >> 5);
  if (row >= n) return;
  const size_t off = (size_t)row * D;
  const float2 a = ((const float2*)(tmp + off))[lane];
  const float ss = wave_reduce_sum(a.x * a.x + a.y * a.y);
  // max(sqrt(ss),1e-12): if norm <= 1e-12 reference divides by 1e-12
  const float inv = (ss > 1e-24f) ? __builtin_amdgcn_rsqf(ss) : 1e12f;
  float2* ap = (float2*)(acc + off) + lane;
  float2 av = *ap;
  av.x += a.x * inv;
  av.y += a.y * inv;
  *ap = av;
  ((float2*)(ebuf + off))[lane] = a;
}

// attv[k] = sum_c att_m[k][c] * att[c]   (folds acc@att_m then *att into acc.attv)
__global__ void attv_kernel(const float* __restrict__ att_m, const float* __restrict__ att,
                            float* __restrict__ attv) {
  const int k = threadIdx.x;
  if (k < D) {
    float s = 0.f;
    for (int c = 0; c < D; ++c) s += att_m[k * D + c] * att[c];
    attv[k] = s;
  }
}

// 2-way softmax over (acc2.attv, acc3.attv); mixed = p2*acc2 + p3*acc3
__global__ void mix_kernel(const float* __restrict__ acc2, const float* __restrict__ acc3,
                           const float* __restrict__ attv, float* __restrict__ mixed,
                           int n) {
  const int lane = threadIdx.x & 31;
  const int row = blockIdx.x * 8 + (threadIdx.x >> 5);
  if (row >= n) return;
  const size_t off = (size_t)row * D;
  const float2 vv = ((const float2*)attv)[lane];
  const float2 a2 = ((const float2*)(acc2 + off))[lane];
  const float2 a3 = ((const float2*)(acc3 + off))[lane];
  const float d2 = wave_reduce_sum(a2.x * vv.x + a2.y * vv.y);
  const float d3 = wave_reduce_sum(a3.x * vv.x + a3.y * vv.y);
  const float mx = fmaxf(d2, d3);
  const float s2 = __expf(d2 - mx), s3 = __expf(d3 - mx);
  const float iz = __builtin_amdgcn_rcpf(s2 + s3);
  const float p2 = s2 * iz, p3 = s3 * iz;
  float2 o;
  o.x = p2 * a2.x + p3 * a3.x;
  o.y = p2 * a2.y + p3 * a3.y;
  ((float2*)(mixed + off))[lane] = o;
}

// Per-batch gather + scaled-dot softmax attention; writes both outputs.
__global__ void final_kernel(const int* __restrict__ input, const float* __restrict__ mixed,
                             float* __restrict__ Lcas, float* __restrict__ cas) {
  __shared__ float s_src[D];
  __shared__ float s_sc[LL];
  const int b = blockIdx.x;
  const int tid = threadIdx.x;
  const int lane = tid & 31, wave = tid >> 5;  // 8 waves of 32
  const int* in_b = input + b * LL;
  const int idx0 = in_b[0];
  if (tid < D) s_src[tid] = mixed[(size_t)idx0 * D + tid];
  __syncthreads();

  for (int l = wave; l < LL; l += 8) {
    const int id = in_b[l];
    const float2 e = ((const float2*)(mixed + (size_t)id * D))[lane];
    const float2 s = ((const float2*)s_src)[lane];
    const float d = wave_reduce_sum(e.x * s.x + e.y * s.y);
    if (lane == 0) s_sc[l] = (id == 0) ? -1e9f : d * 0.125f;  // 1/sqrt(64)
  }
  __syncthreads();

  if (wave == 0) {
    float mx = -1e30f;
    for (int l = lane; l < LL; l += 32) mx = fmaxf(mx, s_sc[l]);
    mx = wave_reduce_max(mx);
    float sm = 0.f;
    for (int l = lane; l < LL; l += 32) sm += __expf(s_sc[l] - mx);
    sm = wave_reduce_sum(sm);
    const float inv = __builtin_amdgcn_rcpf(sm);
    for (int l = lane; l < LL; l += 32) s_sc[l] = __expf(s_sc[l] - mx) * inv;
  }
  __syncthreads();

  for (int l = wave; l < LL; l += 8) {
    const int id = in_b[l];
    const float2 e = ((const float2*)(mixed + (size_t)id * D))[lane];
    const float w = s_sc[l];
    const size_t o = ((size_t)b * LL + l) * D;
    float2 we;
    we.x = w * e.x;
    we.y = w * e.y;
    ((float2*)(cas + o))[lane] = e;
    ((float2*)(Lcas + o))[lane] = we;
  }
}

extern "C" void kernel_launch(void* const* d_in, const int* in_sizes, int n_in,
                              void* d_out, int out_size, void* d_ws, size_t ws_size,
                              hipStream_t stream) {
  const int*   input  = (const int*)d_in[0];
  const int*   it_idx = (const int*)d_in[1];
  const float* it_val = (const float*)d_in[2];
  const int*   us_idx = (const int*)d_in[3];
  const float* us_val = (const float*)d_in[4];
  const float* emb    = (const float*)d_in[5];
  const float* w0     = (const float*)d_in[6];
  const float* w1     = (const float*)d_in[7];
  const float* b0     = (const float*)d_in[8];
  const float* b1     = (const float*)d_in[9];
  const float* att    = (const float*)d_in[10];
  const float* att_m  = (const float*)d_in[11];

  const int n   = in_sizes[5] / D;  // 100000
  const int nnz = in_sizes[2];      // 3200000

  float* ws = (float*)d_ws;
  const size_t ND = (size_t)n * D;
  float* e2   = ws;            // later reused as `mixed`
  float* e3   = ws + ND;
  float* acc2 = ws + 2 * ND;
  float* acc3 = ws + 3 * ND;
  float* tmp  = ws + 4 * ND;
  float* attv = ws + 5 * ND;   // 64 floats

  float* Lcas = (float*)d_out;
  float* cas  = (float*)d_out + (size_t)BB * LL * D;

  gate_kernel<<<(n + 127) / 128, 256, 0, stream>>>(emb, w0, b0, w1, b1,
                                                   e2, acc2, e3, acc3, n);

  // 32 edges per wave, 8 waves per block -> 256 edges per block
  const int spmm_blocks = (nnz + 255) / 256;
  for (int layer = 0; layer < 2; ++layer) {
    hipMemsetAsync(tmp, 0, ND * sizeof(float), stream);
    spmm_kernel<<<spmm_blocks, 256, 0, stream>>>(it_idx, it_val, e2, tmp, nnz);
    norm_acc_kernel<<<(n + 7) / 8, 256, 0, stream>>>(tmp, acc2, e2, n);

    hipMemsetAsync(tmp, 0, ND * sizeof(float), stream);
    spmm_kernel<<<spmm_blocks, 256, 0, stream>>>(us_idx, us_val, e3, tmp, nnz);
    norm_acc_kernel<<<(n + 7) / 8, 256, 0, stream>>>(tmp, acc3, e3, n);
  }

  attv_kernel<<<1, 64, 0, stream>>>(att_m, att, attv);
  mix_kernel<<<(n + 7) / 8, 256, 0, stream>>>(acc2, acc3, attv, /*mixed=*/e2, n);
  final_kernel<<<BB, 256, 0, stream>>>(input, e2, Lcas, cas);
}